// ConstrainNet_43688407335601
// MI455X (gfx1250) — compile-verified
//
#include <hip/hip_runtime.h>
#include <hip/hip_bf16.h>

// ConstrainNet forward on gfx1250 (MI455X), wave32.
//
//   out[0:64]              = V[0, :64] - x0
//   out[(t+1)*64 + s]      = sum_a AB[s][a] * V[t][a]  -  V[t+1][s],  t = 0..126
//
// GEMM: P(128x64, M=t padded) = V(128x96) * ABt(96x64), fp32 via
// V_WMMA_F32_16X16X4_F32 (exact fp32, matches reference precision).
// One wave per 16x16 output tile; 8 M-tiles x 4 N-tiles = 32 waves.

typedef __attribute__((ext_vector_type(2))) float v2f;
typedef __attribute__((ext_vector_type(8))) float v8f;

#define N_STATE 64
#define N_INPUT 32
#define N_ALL   96   // N_STATE + N_INPUT
#define T_STEPS 128

__global__ __launch_bounds__(32)
void constrain_gemm_wmma(const float* __restrict__ A,   // (64,64)
                         const float* __restrict__ B,   // (64,32)
                         const float* __restrict__ V,   // (128,96)
                         float* __restrict__ out)       // (128*64)
{
    const int lane = threadIdx.x;      // 0..31, EXEC all-ones (WMMA requirement)
    const int half = lane >> 4;        // 0: K pair {0,1}; 1: K pair {2,3}
    const int l16  = lane & 15;

    const int m0 = (blockIdx.x >> 2) * 16;   // t-tile base (0..112)
    const int n0 = (blockIdx.x & 3) * 16;    // s-tile base (0..48)

    // A-fragment source: row m of V (lanes 0-15 and 16-31 both cover M=0..15)
    const int m = m0 + l16;
    const float* __restrict__ vrow = V + m * N_ALL;
    // B-fragment source: column n of ABt == row n of [A B]
    const int n = n0 + l16;
    const float* __restrict__ arow = A + n * N_STATE;   // cols 0..63
    const float* __restrict__ brow = B + n * N_INPUT;   // cols 64..95

    v8f acc = {0.f, 0.f, 0.f, 0.f, 0.f, 0.f, 0.f, 0.f};

    #pragma unroll
    for (int k = 0; k < N_ALL; k += 4) {
        const int c = k + half * 2;    // even; never straddles the A/B boundary at 64
        v2f a, b;
        // A-frag (16x4 f32): lane holds V[m][c], V[m][c+1]
        a.x = vrow[c];
        a.y = vrow[c + 1];
        // B-frag (4x16 f32): lane holds ABt[c][n], ABt[c+1][n] = AB[n][c], AB[n][c+1]
        if (c < N_STATE) {
            b.x = arow[c];
            b.y = arow[c + 1];
        } else {
            b.x = brow[c - N_STATE];
            b.y = brow[c - N_STATE + 1];
        }
        // D = A x B + C, fp32, 16x16x4
        acc = __builtin_amdgcn_wmma_f32_16x16x4_f32(
            /*neg_a=*/false, a, /*neg_b=*/false, b,
            /*c_mod=*/(short)0, acc, /*reuse_a=*/false, /*reuse_b=*/false);
    }

    // C/D layout: D[i] -> (M = i + 8*half, N = l16)
    const int s = n0 + l16;
    #pragma unroll
    for (int i = 0; i < 8; ++i) {
        const int t = m0 + i + half * 8;      // GEMM row == timestep t
        if (t < T_STEPS - 1) {                // t = 127 is padding (in-bounds reads, discarded)
            out[(t + 1) * N_STATE + s] = acc[i] - V[(t + 1) * N_ALL + s];
        }
    }
}

__global__ __launch_bounds__(64)
void constrain_row0(const float* __restrict__ V,
                    const float* __restrict__ x0,
                    float* __restrict__ out)
{
    const int s = threadIdx.x;          // 0..63
    out[s] = V[s] - x0[s];
}

extern "C" void kernel_launch(void* const* d_in, const int* in_sizes, int n_in,
                              void* d_out, int out_size, void* d_ws, size_t ws_size,
                              hipStream_t stream)
{
    const float* A  = (const float*)d_in[0];   // (64,64)
    const float* B  = (const float*)d_in[1];   // (64,32)
    const float* x0 = (const float*)d_in[2];   // (64,)
    const float* V  = (const float*)d_in[3];   // net_input, (128*96,)
    float* out = (float*)d_out;                // (8192,)

    (void)in_sizes; (void)n_in; (void)out_size; (void)d_ws; (void)ws_size;

    constrain_row0<<<1, 64, 0, stream>>>(V, x0, out);
    constrain_gemm_wmma<<<32, 32, 0, stream>>>(A, B, V, out);
}